// GINConvNet_66391604462262
// MI455X (gfx1250) — compile-verified
//
#include <hip/hip_runtime.h>
#include <hip/hip_bf16.h>

// ---------------------------------------------------------------------------
// GINConvNet forward on gfx1250 (MI455X).
// All GEMMs use V_WMMA_F32_16X16X4_F32 (wave32, one 16x16 C tile per wave).
// Edge aggregation / pooling use f32 global atomics (L2-resident tables).
// ---------------------------------------------------------------------------

typedef float v2f __attribute__((ext_vector_type(2)));
typedef float v8f __attribute__((ext_vector_type(8)));

#define NN 100000
#define EE 1600000
#define GG 2048
#define NFD 64
#define DH 32
#define SFD 512
#define BN_EPS 1e-5f

// ---- WMMA fragment helpers (CDNA5 ISA §7.12.2 layouts, 32-bit data) -------
// A 16x4 (MxK): lanes 0-15 -> M=lane, {v0,v1}={K+0,K+1}; lanes 16-31 -> {K+2,K+3}
__device__ __forceinline__ v2f load_a_frag(const float* A, int lda, int k) {
    int lane = threadIdx.x & 31;
    int hf = lane >> 4;
    int r  = lane & 15;
    const float* p = A + r * lda + k + 2 * hf;
    v2f f; f.x = p[0]; f.y = p[1];
    return f;
}
// B 4x16 (KxN): mirrored — lanes 0-15 -> N=lane, {v0,v1}={K+0,K+1}; lanes 16-31 -> {K+2,K+3}
__device__ __forceinline__ v2f load_b_frag(const float* B, int ldb, int k) {
    int lane = threadIdx.x & 31;
    int hf = lane >> 4;
    int n  = lane & 15;
    const float* p = B + (k + 2 * hf) * ldb + n;
    v2f f; f.x = p[0]; f.y = p[ldb];
    return f;
}

// ---------------------------------------------------------------------------
// Edge scatter: agg[dst[e]*F + f] += H[src[e]*F + f].  One thread per (e,f);
// consecutive lanes cover consecutive f -> coalesced; tables are L2-resident.
__global__ void edge_scatter_kernel(const int* __restrict__ src,
                                    const int* __restrict__ dst,
                                    const float* __restrict__ H,
                                    float* __restrict__ agg,
                                    int lf, int total) {
    int i = blockIdx.x * blockDim.x + threadIdx.x;
    if (i >= total) return;
    int e = i >> lf;
    int f = i & ((1 << lf) - 1);
    int F = 1 << lf;
    atomicAdd(&agg[dst[e] * F + f], H[src[e] * F + f]);
}

// ---------------------------------------------------------------------------
// Fused GIN MLP: hpre = relu(agg @ Wa + ba) @ Wb + bb  (+ BN column stats).
// One wave (32 threads) per 16-row tile; intermediate 16x32 staged in LDS.
__global__ void gin_mlp_kernel(const float* __restrict__ Agg, int K1,
                               const float* __restrict__ Wa, const float* __restrict__ ba,
                               const float* __restrict__ Wb, const float* __restrict__ bb,
                               float* __restrict__ Hpre,
                               float* __restrict__ colsum, float* __restrict__ colsumsq) {
    __shared__ float h1[16 * DH];
    const int tile = blockIdx.x;
    const float* Ab = Agg + (size_t)tile * 16 * K1;
    const int lane = threadIdx.x;
    const int hf = lane >> 4;
    const int n  = lane & 15;

    v8f acc0 = {0.f,0.f,0.f,0.f,0.f,0.f,0.f,0.f};
    v8f acc1 = {0.f,0.f,0.f,0.f,0.f,0.f,0.f,0.f};
    for (int k = 0; k < K1; k += 4) {
        v2f a  = load_a_frag(Ab, K1, k);
        v2f b0 = load_b_frag(Wa,      DH, k);
        v2f b1 = load_b_frag(Wa + 16, DH, k);
        acc0 = __builtin_amdgcn_wmma_f32_16x16x4_f32(false, a, false, b0, (short)0, acc0, false, false);
        acc1 = __builtin_amdgcn_wmma_f32_16x16x4_f32(false, a, false, b1, (short)0, acc1, false, false);
    }
    // bias + ReLU, stage to LDS (row-major 16x32)
    float bv0 = ba[n], bv1 = ba[16 + n];
    #pragma unroll
    for (int r = 0; r < 8; ++r) {
        int row = r + 8 * hf;
        h1[row * DH + n]      = fmaxf(acc0[r] + bv0, 0.f);
        h1[row * DH + 16 + n] = fmaxf(acc1[r] + bv1, 0.f);
    }
    __syncthreads();

    acc0 = (v8f){0.f,0.f,0.f,0.f,0.f,0.f,0.f,0.f};
    acc1 = (v8f){0.f,0.f,0.f,0.f,0.f,0.f,0.f,0.f};
    for (int k = 0; k < DH; k += 4) {
        v2f a  = load_a_frag(h1, DH, k);
        v2f b0 = load_b_frag(Wb,      DH, k);
        v2f b1 = load_b_frag(Wb + 16, DH, k);
        acc0 = __builtin_amdgcn_wmma_f32_16x16x4_f32(false, a, false, b0, (short)0, acc0, false, false);
        acc1 = __builtin_amdgcn_wmma_f32_16x16x4_f32(false, a, false, b1, (short)0, acc1, false, false);
    }
    float bb0 = bb[n], bb1 = bb[16 + n];
    float s0 = 0.f, q0 = 0.f, s1 = 0.f, q1 = 0.f;
    #pragma unroll
    for (int r = 0; r < 8; ++r) {
        int row = tile * 16 + r + 8 * hf;
        float v0 = acc0[r] + bb0;
        float v1 = acc1[r] + bb1;
        Hpre[(size_t)row * DH + n]      = v0;
        Hpre[(size_t)row * DH + 16 + n] = v1;
        s0 += v0; q0 += v0 * v0;
        s1 += v1; q1 += v1 * v1;
    }
    // combine halves (lane L and L+16 hold the same column)
    s0 += __shfl_xor(s0, 16, 32); q0 += __shfl_xor(q0, 16, 32);
    s1 += __shfl_xor(s1, 16, 32); q1 += __shfl_xor(q1, 16, 32);
    if (hf == 0) {
        atomicAdd(&colsum[n], s0);        atomicAdd(&colsumsq[n], q0);
        atomicAdd(&colsum[16 + n], s1);   atomicAdd(&colsumsq[16 + n], q1);
    }
}

// ---------------------------------------------------------------------------
// BatchNorm (training-mode, biased var) + ReLU, stats from colsum/colsumsq.
__global__ void bn_relu_kernel(const float* __restrict__ Hpre,
                               const float* __restrict__ colsum,
                               const float* __restrict__ colsumsq,
                               const float* __restrict__ g, const float* __restrict__ b,
                               float* __restrict__ H, int total, float invN) {
    int i = blockIdx.x * blockDim.x + threadIdx.x;
    if (i >= total) return;
    int c = i & (DH - 1);
    float mu  = colsum[c] * invN;
    float var = colsumsq[c] * invN - mu * mu;
    float v = (Hpre[i] - mu) * rsqrtf(var + BN_EPS) * g[c] + b[c];
    H[i] = fmaxf(v, 0.f);
}

// ---------------------------------------------------------------------------
// Sum pooling per graph.
__global__ void pool_kernel(const float* __restrict__ H, const int* __restrict__ batch,
                            float* __restrict__ pooled, int total) {
    int i = blockIdx.x * blockDim.x + threadIdx.x;
    if (i >= total) return;
    int node = i >> 5;          // DH == 32
    int c = i & (DH - 1);
    atomicAdd(&pooled[batch[node] * DH + c], H[i]);
}

// ---------------------------------------------------------------------------
// Generic WMMA GEMM: C[m, tn*16+n (with ldc)] = act(A[MxK] @ W[KxNc] + bias)
// One wave per 16x16 tile, 4 waves per block.
__global__ void gemm_wmma_kernel(const float* __restrict__ A, const float* __restrict__ W,
                                 const float* __restrict__ bias, float* __restrict__ C,
                                 int M, int K, int Nc, int ldc, int do_relu) {
    int wave = (blockIdx.x * blockDim.x + threadIdx.x) >> 5;
    int tilesN = Nc >> 4;
    int totalTiles = (M >> 4) * tilesN;
    if (wave >= totalTiles) return;
    int tm = wave / tilesN;
    int tn = wave % tilesN;
    const float* Ab = A + (size_t)tm * 16 * K;
    const float* Bb = W + tn * 16;

    v8f acc = {0.f,0.f,0.f,0.f,0.f,0.f,0.f,0.f};
    for (int k = 0; k < K; k += 4) {
        v2f a = load_a_frag(Ab, K, k);
        v2f b = load_b_frag(Bb, Nc, k);
        acc = __builtin_amdgcn_wmma_f32_16x16x4_f32(false, a, false, b, (short)0, acc, false, false);
    }
    int lane = threadIdx.x & 31;
    int hf = lane >> 4;
    int n  = lane & 15;
    float bv = bias[tn * 16 + n];
    #pragma unroll
    for (int r = 0; r < 8; ++r) {
        float v = acc[r] + bv;
        if (do_relu) v = fmaxf(v, 0.f);
        C[(size_t)(tm * 16 + r + 8 * hf) * ldc + tn * 16 + n] = v;
    }
}

// ---------------------------------------------------------------------------
// Final 256 -> 1 projection: one wave per row.
__global__ void out_kernel(const float* __restrict__ T2, const float* __restrict__ w,
                           const float* __restrict__ b, float* __restrict__ out, int Grows) {
    int wave = (blockIdx.x * blockDim.x + threadIdx.x) >> 5;
    if (wave >= Grows) return;
    int lane = threadIdx.x & 31;
    const float* row = T2 + (size_t)wave * 256;
    float s = 0.f;
    #pragma unroll
    for (int k = lane; k < 256; k += 32) s += row[k] * w[k];
    #pragma unroll
    for (int off = 16; off > 0; off >>= 1) s += __shfl_down(s, off, 32);
    if (lane == 0) out[wave] = s + b[0];
}

// ---------------------------------------------------------------------------
extern "C" void kernel_launch(void* const* d_in, const int* in_sizes, int n_in,
                              void* d_out, int out_size, void* d_ws, size_t ws_size,
                              hipStream_t stream) {
    const float* x    = (const float*)d_in[0];
    const int*   eidx = (const int*)d_in[1];
    const int*   batch= (const int*)d_in[2];
    const float* sf   = (const float*)d_in[3];
    const float* w1a  = (const float*)d_in[4];
    const float* b1a  = (const float*)d_in[5];
    const float* ws_a = (const float*)d_in[6];
    const float* bs_a = (const float*)d_in[7];
    const float* ws_b = (const float*)d_in[8];
    const float* bs_b = (const float*)d_in[9];
    const float* bn_g = (const float*)d_in[10];
    const float* bn_b = (const float*)d_in[11];
    const float* fcg_w= (const float*)d_in[12];
    const float* fcg_b= (const float*)d_in[13];
    const float* fs1_w= (const float*)d_in[14];
    const float* fs1_b= (const float*)d_in[15];
    const float* fs2_w= (const float*)d_in[16];
    const float* fs2_b= (const float*)d_in[17];
    const float* fc1_w= (const float*)d_in[18];
    const float* fc1_b= (const float*)d_in[19];
    const float* fc2_w= (const float*)d_in[20];
    const float* fc2_b= (const float*)d_in[21];
    const float* out_w= (const float*)d_in[22];
    const float* out_b= (const float*)d_in[23];
    float* out = (float*)d_out;

    const int* src = eidx;         // edge_index[0]
    const int* dst = eidx + EE;    // edge_index[1]

    // Workspace layout (floats)
    float* ws = (float*)d_ws;
    float* agg    = ws;                         // N*64
    float* h      = agg    + (size_t)NN * NFD;  // N*32
    float* hpre   = h      + (size_t)NN * DH;   // N*32
    float* colsum = hpre   + (size_t)NN * DH;   // 32
    float* colsq  = colsum + DH;                // 32
    float* pooled = colsq  + DH;                // G*32
    float* z      = pooled + (size_t)GG * DH;   // G*256 (gfeat | s2)
    float* s1     = z      + (size_t)GG * 256;  // G*256
    float* t1     = s1     + (size_t)GG * 256;  // G*1024
    float* t2     = t1     + (size_t)GG * 1024; // G*256

    const int ntiles = NN / 16;                 // 6250
    const float invN = 1.0f / (float)NN;

    for (int l = 0; l < 5; ++l) {
        const float* hin = (l == 0) ? x : h;
        const int F  = (l == 0) ? NFD : DH;
        const int lf = (l == 0) ? 6 : 5;

        // agg = hin (self term), then agg[dst] += hin[src]
        hipMemcpyAsync(agg, hin, (size_t)NN * F * sizeof(float),
                       hipMemcpyDeviceToDevice, stream);
        int etotal = EE * F;
        edge_scatter_kernel<<<(etotal + 255) / 256, 256, 0, stream>>>(
            src, dst, hin, agg, lf, etotal);

        hipMemsetAsync(colsum, 0, 2 * DH * sizeof(float), stream);

        const float* Wa = (l == 0) ? w1a : (ws_a + (size_t)(l - 1) * DH * DH);
        const float* ba = (l == 0) ? b1a : (bs_a + (size_t)(l - 1) * DH);
        const float* Wb = ws_b + (size_t)l * DH * DH;
        const float* bb = bs_b + (size_t)l * DH;
        gin_mlp_kernel<<<ntiles, 32, 0, stream>>>(agg, F, Wa, ba, Wb, bb,
                                                  hpre, colsum, colsq);

        int btotal = NN * DH;
        bn_relu_kernel<<<(btotal + 255) / 256, 256, 0, stream>>>(
            hpre, colsum, colsq, bn_g + (size_t)l * DH, bn_b + (size_t)l * DH,
            h, btotal, invN);
    }

    // ---- pooling ----
    hipMemsetAsync(pooled, 0, (size_t)GG * DH * sizeof(float), stream);
    int ptotal = NN * DH;
    pool_kernel<<<(ptotal + 255) / 256, 256, 0, stream>>>(h, batch, pooled, ptotal);

    // ---- dense head (all WMMA f32) ----
    auto launch_gemm = [&](const float* A, const float* W, const float* bias,
                           float* C, int M, int K, int Nc, int ldc, int relu) {
        int tiles = (M / 16) * (Nc / 16);
        int blocks = (tiles + 3) / 4;           // 4 waves (128 threads) per block
        gemm_wmma_kernel<<<blocks, 128, 0, stream>>>(A, W, bias, C, M, K, Nc, ldc, relu);
    };

    // gfeat = relu(pooled @ fcg_w + fcg_b)  -> z[:, 0:128]
    launch_gemm(pooled, fcg_w, fcg_b, z, GG, DH, 128, 256, 1);
    // s1 = relu(sf @ fs1_w + fs1_b)
    launch_gemm(sf, fs1_w, fs1_b, s1, GG, SFD, 256, 256, 1);
    // s2 = relu(s1 @ fs2_w + fs2_b)         -> z[:, 128:256]
    launch_gemm(s1, fs2_w, fs2_b, z + 128, GG, 256, 128, 256, 1);
    // t1 = relu(z @ fc1_w + fc1_b)
    launch_gemm(z, fc1_w, fc1_b, t1, GG, 256, 1024, 1024, 1);
    // t2 = relu(t1 @ fc2_w + fc2_b)
    launch_gemm(t1, fc2_w, fc2_b, t2, GG, 1024, 256, 256, 1);
    // out = t2 @ out_w + out_b
    out_kernel<<<(GG * 32 + 255) / 256, 256, 0, stream>>>(t2, out_w, out_b, out, GG);
}